// Spatial_Net1_52991306498325
// MI455X (gfx1250) — compile-verified
//
#include <hip/hip_runtime.h>

typedef float v2f __attribute__((ext_vector_type(2)));
typedef float v8f __attribute__((ext_vector_type(8)));

#define N_NODES 85
#define N_EDGES 600
#define F_IN    16
#define F_OUT   20
#define E_TOT   (N_EDGES + N_NODES)   // 685

// GEMM1: [85x16] @ [16x20], padded to M1=96 (6 tiles), K1=16, N1=32 (2 tiles)
#define M1 96
#define K1 16
#define N1 32
// GEMM2: [100x17] @ [17x7], padded to M2=112 (7 tiles), K2=20, N2=16 (1 tile)
#define M2 112
#define K2 20
#define N2 16

__global__ __launch_bounds__(256)
void gcn_fused_kernel(const float* __restrict__ x,
                      const int*   __restrict__ edge_index,
                      const float* __restrict__ Wg,
                      const float* __restrict__ bg,
                      const float* __restrict__ Wf,
                      const float* __restrict__ bf,
                      float*       __restrict__ out)
{
    __shared__ float xpad[M1 * K1];          // 1536 f: padded x
    __shared__ float wgpad[K1 * N1];         //  512 f: padded W_gcn
    __shared__ float xw[M1 * N1];            // 3072 f: x @ W_gcn (padded)
    __shared__ float agg[N_NODES * F_OUT];   // 1700 f: scatter-add target (== x1 flat)
    __shared__ float hpad[M2 * K2];          // 2240 f: relu(reshape) padded
    __shared__ float wfpad[K2 * N2];         //  320 f: padded W_fc
    __shared__ float outb[M2 * N2];          // 1792 f: GEMM2 result (padded)
    __shared__ float dinv[N_NODES];          // degree -> rsqrt(degree)
    __shared__ int   rowL[E_TOT];
    __shared__ int   colL[E_TOT];
    __shared__ float normL[E_TOT];

    const int tid  = threadIdx.x;
    const int lane = tid & 31;
    const int wave = tid >> 5;

    // ---------------- Phase 0: init LDS + stage inputs ----------------
    for (int i = tid; i < M1 * K1; i += 256) xpad[i]  = 0.f;
    for (int i = tid; i < K1 * N1; i += 256) wgpad[i] = 0.f;
    for (int i = tid; i < M2 * K2; i += 256) hpad[i]  = 0.f;
    for (int i = tid; i < K2 * N2; i += 256) wfpad[i] = 0.f;
    for (int i = tid; i < N_NODES * F_OUT; i += 256) agg[i] = bg[i % F_OUT];
    for (int i = tid; i < N_NODES; i += 256) dinv[i] = 0.f;   // used as degree first

    for (int i = tid; i < N_NODES * F_IN; i += 256) {
        int r = i / F_IN, c = i - r * F_IN;
        xpad[r * K1 + c] = x[i];
    }
    for (int i = tid; i < F_IN * F_OUT; i += 256) {
        int r = i / F_OUT, c = i - r * F_OUT;
        wgpad[r * N1 + c] = Wg[i];
    }
    for (int i = tid; i < 17 * 7; i += 256) {
        int r = i / 7, c = i - r * 7;
        wfpad[r * N2 + c] = Wf[i];
    }
    for (int e = tid; e < E_TOT; e += 256) {
        int r, c;
        if (e < N_EDGES) { r = edge_index[e]; c = edge_index[N_EDGES + e]; }
        else             { r = e - N_EDGES;   c = r; }          // self loops
        rowL[e] = r; colL[e] = c;
    }
    __syncthreads();

    // ---------------- Phase 1: degrees -> dinv -> per-edge norm ----------------
    for (int e = tid; e < E_TOT; e += 256) atomicAdd(&dinv[colL[e]], 1.0f);
    __syncthreads();
    for (int i = tid; i < N_NODES; i += 256) {
        float d = dinv[i];
        dinv[i] = (d > 0.f) ? rsqrtf(d) : 0.f;
    }
    __syncthreads();
    for (int e = tid; e < E_TOT; e += 256)
        normL[e] = dinv[rowL[e]] * dinv[colL[e]];

    // ---------------- Phase 2: GEMM1  xw = xpad @ wgpad  (WMMA f32 16x16x4) ----
    // 6 M-tiles x 2 N-tiles = 12 tiles over 8 waves; wave-uniform loop (EXEC all-1s).
    const int mlane = lane & 15;
    const int koff  = (lane >> 4) << 1;        // 0 for lanes 0-15, 2 for lanes 16-31
    for (int t = wave; t < 12; t += 8) {
        const int m0 = (t % 6) * 16;
        const int n0 = (t / 6) * 16;
        const int mr = m0 + mlane;             // A row for this lane
        const int nc = n0 + mlane;             // B/D col for this lane
        v8f acc = {};
        #pragma unroll
        for (int kb = 0; kb < K1; kb += 4) {
            v2f a, b;
            // A 16x4 f32: v0 = K=kb+koff, v1 = K=kb+koff+1
            a.x = xpad[mr * K1 + kb + koff];
            a.y = xpad[mr * K1 + kb + koff + 1];
            // B 4x16 f32: v0 = row K=kb+koff, v1 = row K=kb+koff+1
            b.x = wgpad[(kb + koff)     * N1 + nc];
            b.y = wgpad[(kb + koff + 1) * N1 + nc];
            acc = __builtin_amdgcn_wmma_f32_16x16x4_f32(
                      false, a, false, b, (short)0, acc, false, false);
        }
        // D 16x16 f32: VGPR g -> row m0+g (lanes 0-15) / m0+g+8 (lanes 16-31)
        const int rbase = m0 + ((lane >> 4) << 3);
        #pragma unroll
        for (int g = 0; g < 8; ++g)
            xw[(rbase + g) * N1 + nc] = acc[g];
    }
    __syncthreads();

    // ---------------- Phase 3: gather * norm, LDS scatter-add ----------------
    for (int w = tid; w < E_TOT * F_OUT; w += 256) {
        int e = w / F_OUT, f = w - e * F_OUT;
        atomicAdd(&agg[colL[e] * F_OUT + f], normL[e] * xw[rowL[e] * N1 + f]);
    }
    __syncthreads();

    // ---------------- Phase 4: reshape(100,17) + ReLU into hpad ----------------
    for (int i = tid; i < 100 * 17; i += 256) {
        int r = i / 17, c = i - r * 17;
        float v = agg[i];
        hpad[r * K2 + c] = v > 0.f ? v : 0.f;
    }
    __syncthreads();

    // ---------------- Phase 5: GEMM2  outb = hpad @ wfpad (WMMA f32 16x16x4) ---
    for (int t = wave; t < 7; t += 8) {
        const int m0 = t * 16;
        const int mr = m0 + mlane;
        const int nc = mlane;                  // single N tile
        v8f acc = {};
        #pragma unroll
        for (int kb = 0; kb < K2; kb += 4) {
            v2f a, b;
            a.x = hpad[mr * K2 + kb + koff];
            a.y = hpad[mr * K2 + kb + koff + 1];
            b.x = wfpad[(kb + koff)     * N2 + nc];
            b.y = wfpad[(kb + koff + 1) * N2 + nc];
            acc = __builtin_amdgcn_wmma_f32_16x16x4_f32(
                      false, a, false, b, (short)0, acc, false, false);
        }
        const int rbase = m0 + ((lane >> 4) << 3);
        #pragma unroll
        for (int g = 0; g < 8; ++g)
            outb[(rbase + g) * N2 + nc] = acc[g];
    }
    __syncthreads();

    // ---------------- Phase 6: bias + store [100,7] ----------------
    for (int i = tid; i < 100 * 7; i += 256) {
        int r = i / 7, c = i - r * 7;
        out[i] = outb[r * N2 + c] + bf[c];
    }
}

extern "C" void kernel_launch(void* const* d_in, const int* in_sizes, int n_in,
                              void* d_out, int out_size, void* d_ws, size_t ws_size,
                              hipStream_t stream) {
    const float* x    = (const float*)d_in[0];   // [85,16]
    const int*   ei   = (const int*)d_in[1];     // [2,600]
    const float* Wg   = (const float*)d_in[2];   // [16,20]
    const float* bg   = (const float*)d_in[3];   // [20]
    const float* Wf   = (const float*)d_in[4];   // [17,7]
    const float* bf   = (const float*)d_in[5];   // [7]
    float*       out  = (float*)d_out;           // [100,7]
    (void)in_sizes; (void)n_in; (void)out_size; (void)d_ws; (void)ws_size;

    gcn_fused_kernel<<<1, 256, 0, stream>>>(x, ei, Wg, bg, Wf, bf, out);
}